// Encoder_29824252903864
// MI455X (gfx1250) — compile-verified
//
#include <hip/hip_runtime.h>
#include <math.h>

// ---------------------------------------------------------------------------
// Transformer encoder layer for MI455X (gfx1250).
// fp32 WMMA (V_WMMA_F32_16X16X4_F32) + Tensor Data Mover (TDM) global->LDS
// staging with double buffering, synchronized via TENSORcnt.
// B=8, S=512, I=1024, H=16, P=64, U=4096.  BS = 4096 rows.
// ---------------------------------------------------------------------------

typedef float v2f __attribute__((ext_vector_type(2)));
typedef float v8f __attribute__((ext_vector_type(8)));
typedef unsigned int v4u __attribute__((ext_vector_type(4)));
typedef int v8i __attribute__((ext_vector_type(8)));
typedef int v4i __attribute__((ext_vector_type(4)));

#define LN_EPS 1e-5f

__device__ __forceinline__ v8f wmma_f32(v2f a, v2f b, v8f c) {
  return __builtin_amdgcn_wmma_f32_16x16x4_f32(false, a, false, b,
                                               (short)0, c, false, false);
}

__device__ __forceinline__ float gelu_exact(float x) {
  return 0.5f * x * (1.0f + erff(x * 0.70710678118654752440f));
}

// ---------------------------------------------------------------------------
// TDM: DMA a 2D tile of f32 (tile0 x tile1, row pitch stride_elems in memory)
// from global into LDS at lds_addr, inserting 1 DWORD of LDS padding after
// every row (pad_code: 3 -> 16 dwords/row, 5 -> 64, 6 -> 128).
// Tensor dims are set equal to tile dims (tile fully in-bounds by design).
// D# packing per CDNA5 ISA 8.3/8.4 (group0 128b, group1 256b).
// This toolchain uses the 6-arg builtin:
//   (uint32x4 g0, int32x8 g1, int32x4 g2, int32x4 g3, int32x8 x, i32 cpol)
// ---------------------------------------------------------------------------
__device__ __forceinline__ void tdm_load_2d(unsigned lds_addr, const void* gp,
                                            unsigned tile0, unsigned tile1,
                                            unsigned stride_elems,
                                            unsigned pad_code) {
  unsigned long long ga = (unsigned long long)(uintptr_t)gp;
  v4u g0;
  g0.x = 1u;                                   // count=1 (valid), user mode
  g0.y = lds_addr;                             // LDS byte address
  g0.z = (unsigned)(ga & 0xFFFFFFFFu);         // global_addr[31:0]
  g0.w = (unsigned)((ga >> 32) & 0x1FFFFFFu)   // global_addr[56:32]
         | (2u << 30);                         // type = 2 ("image")
  v8i g1;
  g1[0] = (int)((2u << 16)                     // data_size = 4B
                | (1u << 20)                   // pad_enable
                | (pad_code << 22));           // pad_interval; pad_amount=1dw
  g1[1] = (int)((tile0 & 0xFFFFu) << 16);      // tensor_dim0[15:0]
  g1[2] = (int)(((tile0 >> 16) & 0xFFFFu)      // tensor_dim0[31:16]
                | ((tile1 & 0xFFFFu) << 16));  // tensor_dim1[15:0]
  g1[3] = (int)(((tile1 >> 16) & 0xFFFFu)      // tensor_dim1[31:16]
                | ((tile0 & 0xFFFFu) << 16));  // tile_dim0
  g1[4] = (int)(tile1 & 0xFFFFu);              // tile_dim1 (tile_dim2 = 0)
  g1[5] = (int)stride_elems;                   // tensor_dim0_stride[31:0]
  g1[6] = 0;                                   // stride[47:32], dim1_stride lo
  g1[7] = 0;
  v4i z4 = {0, 0, 0, 0};
  v8i z8 = {0, 0, 0, 0, 0, 0, 0, 0};
  __builtin_amdgcn_tensor_load_to_lds(g0, g1, z4, z4, z8, 0);
}

__device__ __forceinline__ unsigned lds_off(const void* p) {
  return (unsigned)(uintptr_t)p;  // low 32 bits of generic addr = LDS offset
}

// ---------------------------------------------------------------------------
// Tiled GEMM: C[M,N] = A[M,K] * B[K,N] + bias[N]  (optional exact GELU)
// Block: 256 threads (8 waves), tile 128x128. Wave tile 32x64 (2x4 accs).
// TDM loads K-chunks of 16 into double-buffered LDS; wave 0 drives the DMA.
// Requires M%128==0, N%128==0, K%16==0.
// ---------------------------------------------------------------------------
template <int GELU>
__global__ __launch_bounds__(256) void gemm_bias_kernel(
    const float* __restrict__ A, const float* __restrict__ B,
    const float* __restrict__ bias, float* __restrict__ C,
    int M, int N, int K) {
  __shared__ float sA[2][128][17];  // [buf][row][k], TDM-padded pitch 17
  __shared__ float sB[2][16][129];  // [buf][k][col], TDM-padded pitch 129

  const int tid  = threadIdx.x;
  const int wave = tid >> 5;
  const int lane = tid & 31;
  const int wm   = (wave & 3) * 32;  // wave row offset (4 waves down)
  const int wn   = (wave >> 2) * 64; // wave col offset (2 waves across)
  const int l15  = lane & 15;
  const int kh   = lane >> 4;

  const int row0 = blockIdx.y * 128;
  const int col0 = blockIdx.x * 128;
  const bool loader = (wave == 0);

  v8f acc[2][4] = {};

  // Prologue: DMA chunk 0 into buffer 0.
  if (loader) {
    tdm_load_2d(lds_off(&sA[0][0][0]), &A[(size_t)row0 * K], 16, 128, K, 3);
    tdm_load_2d(lds_off(&sB[0][0][0]), &B[col0], 128, 16, N, 6);
    __builtin_amdgcn_s_wait_tensorcnt(0);
  }
  __syncthreads();

  int buf = 0;
  for (int k0 = 0; k0 < K; k0 += 16) {
    const int nbuf = buf ^ 1;
    // Kick off DMA for the next chunk while this one is consumed.
    if (loader && (k0 + 16 < K)) {
      tdm_load_2d(lds_off(&sA[nbuf][0][0]),
                  &A[(size_t)row0 * K + (k0 + 16)], 16, 128, K, 3);
      tdm_load_2d(lds_off(&sB[nbuf][0][0]),
                  &B[(size_t)(k0 + 16) * N + col0], 128, 16, N, 6);
    }

#pragma unroll
    for (int kk = 0; kk < 16; kk += 4) {
      v2f af[2], bf[4];
#pragma unroll
      for (int t = 0; t < 2; ++t) {
        af[t].x = sA[buf][wm + t * 16 + l15][kk + 2 * kh];
        af[t].y = sA[buf][wm + t * 16 + l15][kk + 2 * kh + 1];
      }
#pragma unroll
      for (int u = 0; u < 4; ++u) {
        bf[u].x = sB[buf][kk + 2 * kh][wn + u * 16 + l15];
        bf[u].y = sB[buf][kk + 2 * kh + 1][wn + u * 16 + l15];
      }
#pragma unroll
      for (int t = 0; t < 2; ++t)
#pragma unroll
        for (int u = 0; u < 4; ++u)
          acc[t][u] = wmma_f32(af[t], bf[u], acc[t][u]);
    }

    if (loader) __builtin_amdgcn_s_wait_tensorcnt(0);
    __syncthreads();
    buf = nbuf;
  }

  // Epilogue: bias (+ GELU). C/D layout: vgpr r -> row r + 8*kh, col l15.
#pragma unroll
  for (int t = 0; t < 2; ++t)
#pragma unroll
    for (int u = 0; u < 4; ++u)
#pragma unroll
      for (int r = 0; r < 8; ++r) {
        int gr = row0 + wm + t * 16 + r + 8 * kh;
        int gc = col0 + wn + u * 16 + l15;
        float v = acc[t][u][r] + bias[gc];
        if (GELU) v = gelu_exact(v);
        C[(size_t)gr * N + gc] = v;
      }
}

// ---------------------------------------------------------------------------
// Fused attention: one block per (b, h, 64-row q tile). 128 threads (4 waves).
// Q/K/V 64x64 tiles DMA'd by TDM (stride 1024, pad code 5 -> pitch 65).
// scores = (Q K^T)/8 held as full 64x512 block in LDS, softmax, then P*V.
// ---------------------------------------------------------------------------
__global__ __launch_bounds__(128) void attention_kernel(
    const float* __restrict__ Q, const float* __restrict__ Km,
    const float* __restrict__ V, float* __restrict__ O) {
  extern __shared__ float smem[];
  float* sQ = smem;              // 64 x 65
  float* sK = sQ + 64 * 65;      // 64 x 65
  float* sV = sK + 64 * 65;      // 64 x 65
  float* sS = sV + 64 * 65;      // 64 x 520

  const int q0   = blockIdx.x * 64;
  const int h    = blockIdx.y;
  const int b    = blockIdx.z;
  const int tid  = threadIdx.x;
  const int wave = tid >> 5;
  const int lane = tid & 31;
  const int l15  = lane & 15;
  const int kh   = lane >> 4;
  const size_t base = (size_t)b * 512 * 1024 + (size_t)h * 64;
  const bool loader = (wave == 0);

  // DMA the Q tile.
  if (loader)
    tdm_load_2d(lds_off(sQ), &Q[base + (size_t)q0 * 1024], 64, 64, 1024, 5);

  // ---- scores: for each 64-key tile, WMMA over P=64 ----
  for (int kt = 0; kt < 8; ++kt) {
    __syncthreads();  // previous sK consumers done
    if (loader) {
      tdm_load_2d(lds_off(sK), &Km[base + (size_t)(kt * 64) * 1024],
                  64, 64, 1024, 5);
      __builtin_amdgcn_s_wait_tensorcnt(0);  // also covers Q on kt==0
    }
    __syncthreads();

    v8f acc[4] = {};
#pragma unroll
    for (int kp = 0; kp < 64; kp += 4) {
      v2f bf;  // B = K^T: [p][key]; key = wave*16 + l15
      bf.x = sK[(wave * 16 + l15) * 65 + kp + 2 * kh];
      bf.y = sK[(wave * 16 + l15) * 65 + kp + 2 * kh + 1];
#pragma unroll
      for (int t = 0; t < 4; ++t) {
        v2f af;
        af.x = sQ[(t * 16 + l15) * 65 + kp + 2 * kh];
        af.y = sQ[(t * 16 + l15) * 65 + kp + 2 * kh + 1];
        acc[t] = wmma_f32(af, bf, acc[t]);
      }
    }
#pragma unroll
    for (int t = 0; t < 4; ++t)
#pragma unroll
      for (int r = 0; r < 8; ++r)
        sS[(t * 16 + r + 8 * kh) * 520 + kt * 64 + wave * 16 + l15] =
            acc[t][r] * 0.125f;  // 1/sqrt(64)
  }
  __syncthreads();

  // ---- softmax over 512-wide rows (threads 0..63, one row each) ----
  if (tid < 64) {
    float* row = &sS[tid * 520];
    float mx = -3.4e38f;
    for (int j = 0; j < 512; ++j) mx = fmaxf(mx, row[j]);
    float sum = 0.f;
    for (int j = 0; j < 512; ++j) {
      float e = __expf(row[j] - mx);
      row[j] = e;
      sum += e;
    }
    float inv = 1.0f / sum;
    for (int j = 0; j < 512; ++j) row[j] *= inv;
  }
  __syncthreads();

  // ---- ctx = probs (64x512) x V (512x64) ----
  v8f acc[4] = {};
  for (int kt = 0; kt < 8; ++kt) {
    if (loader) {
      tdm_load_2d(lds_off(sV), &V[base + (size_t)(kt * 64) * 1024],
                  64, 64, 1024, 5);
      __builtin_amdgcn_s_wait_tensorcnt(0);
    }
    __syncthreads();
#pragma unroll
    for (int kk = 0; kk < 64; kk += 4) {
      v2f bf;  // B = V: [key][p]; out col p = wave*16 + l15
      bf.x = sV[(kk + 2 * kh) * 65 + wave * 16 + l15];
      bf.y = sV[(kk + 2 * kh + 1) * 65 + wave * 16 + l15];
#pragma unroll
      for (int t = 0; t < 4; ++t) {
        v2f af;
        af.x = sS[(t * 16 + l15) * 520 + kt * 64 + kk + 2 * kh];
        af.y = sS[(t * 16 + l15) * 520 + kt * 64 + kk + 2 * kh + 1];
        acc[t] = wmma_f32(af, bf, acc[t]);
      }
    }
    __syncthreads();  // all waves done with sV before next DMA
  }

  // Write ctx back as [b][q][h*64+p]
#pragma unroll
  for (int t = 0; t < 4; ++t)
#pragma unroll
    for (int r = 0; r < 8; ++r) {
      int lr = t * 16 + r + 8 * kh;
      int lc = wave * 16 + l15;
      O[base + (size_t)(q0 + lr) * 1024 + lc] = acc[t][r];
    }
}

// ---------------------------------------------------------------------------
// out[row] = LayerNorm(A[row] + R[row]) * scale + bias, I = 1024.
// ---------------------------------------------------------------------------
__global__ __launch_bounds__(256) void ln_residual_kernel(
    const float* __restrict__ A, const float* __restrict__ R,
    const float* __restrict__ scale, const float* __restrict__ bias,
    float* __restrict__ out) {
  __shared__ float red[256];
  const int row = blockIdx.x;
  const int tid = threadIdx.x;
  const size_t off = (size_t)row * 1024;

  float v[4];
  float s = 0.f;
#pragma unroll
  for (int j = 0; j < 4; ++j) {
    int c = tid + j * 256;
    v[j] = A[off + c] + R[off + c];
    s += v[j];
  }
  red[tid] = s;
  __syncthreads();
  for (int k = 128; k > 0; k >>= 1) {
    if (tid < k) red[tid] += red[tid + k];
    __syncthreads();
  }
  float mu = red[0] * (1.0f / 1024.0f);
  __syncthreads();

  s = 0.f;
#pragma unroll
  for (int j = 0; j < 4; ++j) {
    float d = v[j] - mu;
    s += d * d;
  }
  red[tid] = s;
  __syncthreads();
  for (int k = 128; k > 0; k >>= 1) {
    if (tid < k) red[tid] += red[tid + k];
    __syncthreads();
  }
  float rstd = rsqrtf(red[0] * (1.0f / 1024.0f) + LN_EPS);

#pragma unroll
  for (int j = 0; j < 4; ++j) {
    int c = tid + j * 256;
    out[off + c] = (v[j] - mu) * rstd * scale[c] + bias[c];
  }
}

// ---------------------------------------------------------------------------
extern "C" void kernel_launch(void* const* d_in, const int* in_sizes, int n_in,
                              void* d_out, int out_size, void* d_ws,
                              size_t ws_size, hipStream_t stream) {
  (void)in_sizes; (void)n_in; (void)out_size; (void)ws_size;

  const float* x      = (const float*)d_in[0];
  const float* w_qkv  = (const float*)d_in[1];
  const float* b_qkv  = (const float*)d_in[2];
  const float* w_out  = (const float*)d_in[3];
  const float* b_out  = (const float*)d_in[4];
  const float* ln1_s  = (const float*)d_in[5];
  const float* ln1_b  = (const float*)d_in[6];
  const float* w_ffn1 = (const float*)d_in[7];
  const float* b_ffn1 = (const float*)d_in[8];
  const float* w_ffn2 = (const float*)d_in[9];
  const float* b_ffn2 = (const float*)d_in[10];
  const float* ln2_s  = (const float*)d_in[11];
  const float* ln2_b  = (const float*)d_in[12];
  float* out = (float*)d_out;
  float* ws  = (float*)d_ws;

  const int BS = 4096, I = 1024, U = 4096;
  const size_t SZ = (size_t)BS * I;  // 4M floats

  // Workspace layout (peak 28M floats = 112 MB):
  float* qkv  = ws;           // [0, 3SZ): q,k,v — live through attention
  float* ctx  = ws + 3 * SZ;  // [3SZ, 4SZ): live through out-proj
  float* attn = ws;           // reuse q region after attention
  float* r1   = ws + SZ;      // reuse k region
  float* ffn  = ws + 2 * SZ;  // reuse v region
  float* hbuf = ws + 3 * SZ;  // [3SZ, 7SZ): overwrites ctx after out-proj

  dim3 blk(256);

  // 1) QKV projection: qkv[t] = x @ w_qkv[t] + b_qkv[t]
  for (int t = 0; t < 3; ++t) {
    gemm_bias_kernel<0><<<dim3(I / 128, BS / 128), blk, 0, stream>>>(
        x, w_qkv + (size_t)t * I * I, b_qkv + (size_t)t * I,
        qkv + (size_t)t * SZ, BS, I, I);
  }

  // 2) Attention (B=8, H=16, 8 q-tiles of 64)
  size_t attn_lds = (size_t)(3 * 64 * 65 + 64 * 520) * sizeof(float);
  attention_kernel<<<dim3(8, 16, 8), dim3(128), attn_lds, stream>>>(
      qkv, qkv + SZ, qkv + 2 * SZ, ctx);

  // 3) Output projection
  gemm_bias_kernel<0><<<dim3(I / 128, BS / 128), blk, 0, stream>>>(
      ctx, w_out, b_out, attn, BS, I, I);

  // 4) Residual 1 + LayerNorm
  ln_residual_kernel<<<BS, 256, 0, stream>>>(x, attn, ln1_s, ln1_b, r1);

  // 5) FFN1 + exact GELU
  gemm_bias_kernel<1><<<dim3(U / 128, BS / 128), blk, 0, stream>>>(
      r1, w_ffn1, b_ffn1, hbuf, BS, U, I);

  // 6) FFN2
  gemm_bias_kernel<0><<<dim3(I / 128, BS / 128), blk, 0, stream>>>(
      hbuf, w_ffn2, b_ffn2, ffn, BS, I, U);

  // 7) Residual 2 + LayerNorm -> output
  ln_residual_kernel<<<BS, 256, 0, stream>>>(r1, ffn, ln2_s, ln2_b, out);
}